// SelectiveSSM_43602507989585
// MI455X (gfx1250) — compile-verified
//
#include <hip/hip_runtime.h>
#include <hip/hip_bf16.h>

// SelectiveSSM for MI455X (gfx1250, wave32).
// Pipeline:
//   K1 : U = (alpha (x) x) @ Bflat                       [WMMA f32 16x16x4]
//   kA : per-(batch,chunk) compose affine map (M_c, v_c) [WMMA f32 16x16x4]
//   kB : 64-step chunk-level scan  h <- h*M_c + v_c      [tiny serial]
//   kC : parallel replay of each chunk -> full h history
//   K3 : y = (alpha (x) x) @ Dflat + (alpha (x) h) @ Cflat [WMMA f32 16x16x4]
// The mixture is linear in alpha, so all bilinear terms hoist out of the
// recurrence; the chunked scan cuts the serial critical path 4096 -> ~128.

typedef float v2f __attribute__((ext_vector_type(2)));
typedef float v8f __attribute__((ext_vector_type(8)));

#define B_  16
#define T_  4096
#define D_  256
#define N_  64
#define K_  4
#define BT_ (B_ * T_)   // 65536
#define L_  64          // chunk length
#define C_  (T_ / L_)   // 64 chunks

static __device__ __forceinline__ v8f wmma4(v2f a, v2f b, v8f c) {
    return __builtin_amdgcn_wmma_f32_16x16x4_f32(false, a, false, b,
                                                 (short)0, c, false, false);
}

// ---------------------------------------------------------------------------
// K1: U[bt,n] = sum_{k,d} (alpha[bt,k]*x[bt,d]) * Bs[k][n][d]
//     M=65536, N=64, K=1024; A generated on the fly.
// ---------------------------------------------------------------------------
__global__ __launch_bounds__(128)
void k1_xbmix(const float* __restrict__ x, const float* __restrict__ alpha,
              const float* __restrict__ Bs, float* __restrict__ U) {
    __shared__ float xs[16 * 257];
    __shared__ float al[16 * 4];
    const int tid = threadIdx.x;
    const int bt0 = blockIdx.x * 16;

    for (int i = tid; i < 16 * 256; i += 128) {
        int r = i >> 8, c = i & 255;
        xs[r * 257 + c] = x[(bt0 + r) * 256 + c];
    }
    if (tid < 64) al[tid] = alpha[bt0 * 4 + tid];
    __syncthreads();

    const int lane = tid & 31, wave = tid >> 5;
    const int n0 = wave * 16;
    const int r = lane & 15, half = lane >> 4, koff = half * 2;
    const float* xrow = &xs[r * 257];
    const float* arow = &al[r * 4];
    const int n = n0 + r;

    v8f acc = {};
    for (int kb = 0; kb < 1024; kb += 4) {
        int   k0 = kb + koff;
        int   kk = k0 >> 8;
        int   d  = k0 & 255;
        float am = arow[kk];
        v2f a, b;
        a.x = am * xrow[d];
        a.y = am * xrow[d + 1];
        const float* bp = &Bs[(kk * 64 + n) * 256 + d];
        b.x = bp[0];
        b.y = bp[1];
        acc = wmma4(a, b, acc);
    }
#pragma unroll
    for (int rr = 0; rr < 8; ++rr)
        U[(bt0 + rr + half * 8) * 64 + n] = acc[rr];
}

// ---------------------------------------------------------------------------
// kA: compose chunk affine map. One block per (b, chunk): 1024 blocks x 128.
//   per step s: Amix_t = sum_k alpha_k As[k];  M <- M*Amix (WMMA);
//               v <- v*Amix + U[t]
// LDS: Assh 64K | Msh 16.6K | AT 16.6K -> ~100 KiB (fits 320 KiB WGP LDS).
// ---------------------------------------------------------------------------
__global__ __launch_bounds__(128)
void kA_compose(const float* __restrict__ alpha, const float* __restrict__ As,
                const float* __restrict__ U, float* __restrict__ Mall,
                float* __restrict__ vall) {
    __shared__ float Assh[K_ * N_ * N_];   // Aflat[kn][m] (as stored)
    __shared__ float Msh[N_ * 65];         // M, padded rows
    __shared__ float AT[N_ * 65];          // Amix^T : AT[m][n]
    __shared__ float vsh[N_];
    __shared__ float psum[2 * N_];

    const int tid = threadIdx.x;
    const int bc  = blockIdx.x;            // b*C_ + c
    const int b   = bc >> 6;
    const int c   = bc & 63;

    for (int i = tid; i < K_ * N_ * N_; i += 128) Assh[i] = As[i];
    for (int i = tid; i < N_ * N_; i += 128) {
        int nn = i >> 6, mm = i & 63;
        Msh[nn * 65 + mm] = (nn == mm) ? 1.0f : 0.0f;
    }
    if (tid < N_) vsh[tid] = 0.0f;
    __syncthreads();

    const int lane = tid & 31, wave = tid >> 5;
    const int r = lane & 15, half = lane >> 4, koff = half * 2;
    const int vm = tid & 63, vpart = tid >> 6;   // v-GEMV: 2 parts x 32 rows
    const long t0 = (long)b * T_ + (long)c * L_;

    for (int s = 0; s < L_; ++s) {
        const long bt = t0 + s;
        const float4 a4 = ((const float4*)alpha)[bt];   // 16B-aligned
        // build Amix^T
        for (int i = tid; i < N_ * N_; i += 128) {
            int nn = i >> 6, mm = i & 63;
            float v = a4.x * Assh[(0 * 64 + nn) * 64 + mm]
                    + a4.y * Assh[(1 * 64 + nn) * 64 + mm]
                    + a4.z * Assh[(2 * 64 + nn) * 64 + mm]
                    + a4.w * Assh[(3 * 64 + nn) * 64 + mm];
            AT[mm * 65 + nn] = v;
        }
        __syncthreads();    // AT ready; Msh/vsh from previous step ready

        // M*Amix: wave owns result rows [16*wave, 16*wave+16)
        v8f acc[4] = {v8f{}, v8f{}, v8f{}, v8f{}};
        const float* arow = &Msh[(16 * wave + r) * 65];
        for (int kb = 0; kb < 64; kb += 4) {
            v2f a;
            a.x = arow[kb + koff];
            a.y = arow[kb + koff + 1];
#pragma unroll
            for (int J = 0; J < 4; ++J) {
                v2f bf;
                const float* bp = &AT[(16 * J + r) * 65 + kb + koff];
                bf.x = bp[0];
                bf.y = bp[1];
                acc[J] = wmma4(a, bf, acc[J]);
            }
        }
        // v*Amix partials (uses AT, vsh)
        {
            float local = 0.0f;
            const int nb = vpart * 32;
#pragma unroll 8
            for (int j = 0; j < 32; ++j)
                local = __builtin_fmaf(vsh[nb + j], AT[vm * 65 + nb + j], local);
            psum[vpart * 64 + vm] = local;
        }
        __syncthreads();    // all Msh/vsh/AT reads complete; psum ready

        // commit M and v for next step
#pragma unroll
        for (int J = 0; J < 4; ++J)
#pragma unroll
            for (int rr = 0; rr < 8; ++rr)
                Msh[(16 * wave + rr + 8 * half) * 65 + 16 * J + r] = acc[J][rr];
        if (tid < 64)
            vsh[tid] = psum[tid] + psum[64 + tid] + U[bt * 64 + tid];
        // next iteration's first barrier publishes Msh/vsh
    }
    __syncthreads();
    for (int i = tid; i < N_ * N_; i += 128)
        Mall[(long)bc * 4096 + i] = Msh[(i >> 6) * 65 + (i & 63)];
    if (tid < 64) vall[(long)bc * 64 + tid] = vsh[tid];
}

// ---------------------------------------------------------------------------
// kB: chunk-level scan, one block per batch (16 x 256 threads):
//     hstart[c] = h;  h <- h*M_c + v_c       (64 serial chunk steps)
// ---------------------------------------------------------------------------
__global__ __launch_bounds__(256)
void kB_chunkscan(const float* __restrict__ Mall, const float* __restrict__ vall,
                  float* __restrict__ hstart, float* __restrict__ hT) {
    __shared__ float h[64];
    __shared__ float psum[4 * 64];
    const int tid = threadIdx.x, b = blockIdx.x;
    const int m = tid & 63, part = tid >> 6;   // 4 parts x 16 rows
    if (tid < 64) h[tid] = 0.0f;
    __syncthreads();
    for (int c = 0; c < C_; ++c) {
        const long bc = (long)b * C_ + c;
        if (tid < 64) hstart[bc * 64 + tid] = h[tid];
        float local = 0.0f;
        const float* Mc = &Mall[bc * 4096];
        const int n0 = part * 16;
#pragma unroll
        for (int j = 0; j < 16; ++j)
            local = __builtin_fmaf(h[n0 + j], Mc[(n0 + j) * 64 + m], local);
        psum[part * 64 + m] = local;
        __syncthreads();
        if (tid < 64)
            h[tid] = psum[tid] + psum[64 + tid] + psum[128 + tid] +
                     psum[192 + tid] + vall[bc * 64 + tid];
        __syncthreads();
    }
    if (tid < 64) hT[b * 64 + tid] = h[tid];
}

// ---------------------------------------------------------------------------
// kC: parallel replay. One block per (b, chunk): 1024 blocks x 256 threads.
//     h <- (alpha (x) h) @ Aflat + U[t]; emit hist[t] = h for all 64 steps.
// ---------------------------------------------------------------------------
__global__ __launch_bounds__(256)
void kC_replay(const float* __restrict__ alpha, const float* __restrict__ As,
               const float* __restrict__ U, const float* __restrict__ hstart,
               float* __restrict__ hist) {
    __shared__ float Af[256 * 64];
    __shared__ float ha[256];
    __shared__ float h[64];
    __shared__ float psum[4 * 64];
    const int tid = threadIdx.x;
    const int bc  = blockIdx.x;
    for (int i = tid; i < 256 * 64; i += 256) Af[i] = As[i];
    if (tid < 64) h[tid] = hstart[(long)bc * 64 + tid];
    __syncthreads();

    const int m = tid & 63, part = tid >> 6;   // 4 parts x 64 rows
    const int b = bc >> 6, c = bc & 63;
    const long t0 = (long)b * T_ + (long)c * L_;

    for (int s = 0; s < L_; ++s) {
        const long bt = t0 + s;
        ha[tid] = alpha[bt * 4 + (tid >> 6)] * h[tid & 63];
        __syncthreads();
        float local = 0.0f;
        const int row0 = part * 64;
#pragma unroll 8
        for (int j = 0; j < 64; ++j)
            local = __builtin_fmaf(ha[row0 + j], Af[(row0 + j) * 64 + m], local);
        psum[part * 64 + m] = local;
        __syncthreads();
        if (tid < 64) {
            float hn = U[bt * 64 + tid] + psum[tid] + psum[64 + tid] +
                       psum[128 + tid] + psum[192 + tid];
            h[tid] = hn;
            hist[bt * 64 + tid] = hn;
        }
        __syncthreads();
    }
}

// ---------------------------------------------------------------------------
// K3: y[bt,e] = sum_{k,d}(a*x)[kd]*Ds[k][e][d] + sum_{k,n}(a*h)[kn]*Cs[k][e][n]
//     Fused GEMM M=65536, N=256, K=1280.
// ---------------------------------------------------------------------------
__global__ __launch_bounds__(128)
void k3_y(const float* __restrict__ x, const float* __restrict__ alpha,
          const float* __restrict__ Cs, const float* __restrict__ Ds,
          const float* __restrict__ hist, float* __restrict__ y) {
    __shared__ float xs[16 * 257];
    __shared__ float hs[16 * 65];
    __shared__ float al[16 * 4];
    const int tid = threadIdx.x;
    const int bt0 = blockIdx.x * 16;

    for (int i = tid; i < 16 * 256; i += 128) {
        int r = i >> 8, c = i & 255;
        xs[r * 257 + c] = x[(bt0 + r) * 256 + c];
    }
    for (int i = tid; i < 16 * 64; i += 128) {
        int r = i >> 6, c = i & 63;
        hs[r * 65 + c] = hist[(bt0 + r) * 64 + c];
    }
    if (tid < 64) al[tid] = alpha[bt0 * 4 + tid];
    __syncthreads();

    const int lane = tid & 31, wave = tid >> 5;
    const int n0 = (blockIdx.y * 4 + wave) * 16;
    const int r = lane & 15, half = lane >> 4, koff = half * 2;
    const float* xrow = &xs[r * 257];
    const float* hrow = &hs[r * 65];
    const float* arow = &al[r * 4];
    const int e = n0 + r;

    v8f acc = {};
    for (int kb = 0; kb < 1024; kb += 4) {      // xa @ Dflat
        int   k0 = kb + koff;
        int   kk = k0 >> 8;
        int   d  = k0 & 255;
        float am = arow[kk];
        v2f a, b;
        a.x = am * xrow[d];
        a.y = am * xrow[d + 1];
        const float* bp = &Ds[kk * 65536 + e * 256 + d];
        b.x = bp[0];
        b.y = bp[1];
        acc = wmma4(a, b, acc);
    }
    for (int kb = 0; kb < 256; kb += 4) {       // (alpha (x) h) @ Cflat
        int   k0 = kb + koff;
        int   kk = k0 >> 6;
        int   n  = k0 & 63;
        float am = arow[kk];
        v2f a, b;
        a.x = am * hrow[n];
        a.y = am * hrow[n + 1];
        const float* cp = &Cs[kk * 16384 + e * 64 + n];
        b.x = cp[0];
        b.y = cp[1];
        acc = wmma4(a, b, acc);
    }
#pragma unroll
    for (int rr = 0; rr < 8; ++rr)
        y[(size_t)(bt0 + rr + half * 8) * 256 + e] = acc[rr];
}

// ---------------------------------------------------------------------------
extern "C" void kernel_launch(void* const* d_in, const int* in_sizes, int n_in,
                              void* d_out, int out_size, void* d_ws, size_t ws_size,
                              hipStream_t stream) {
    const float* x     = (const float*)d_in[0];   // (B,T,D)
    const float* alpha = (const float*)d_in[1];   // (B,T,K)
    const float* As    = (const float*)d_in[2];   // (K,N,N)
    const float* Bs    = (const float*)d_in[3];   // (K,N,D)
    const float* Cs    = (const float*)d_in[4];   // (K,D,N)
    const float* Ds    = (const float*)d_in[5];   // (K,D,D)

    float* y  = (float*)d_out;                    // (B,T,D) flat
    float* hT = y + (size_t)BT_ * D_;             // (B,N) appended

    // workspace layout (~48.5 MiB total):
    float* U      = (float*)d_ws;                         // BT*64
    float* hist   = U      + (size_t)BT_ * N_;            // BT*64
    float* Mall   = hist   + (size_t)BT_ * N_;            // B*C*64*64
    float* vall   = Mall   + (size_t)B_ * C_ * N_ * N_;   // B*C*64
    float* hstart = vall   + (size_t)B_ * C_ * N_;        // B*C*64

    k1_xbmix   <<<dim3(BT_ / 16),    128, 0, stream>>>(x, alpha, Bs, U);
    kA_compose <<<dim3(B_ * C_),     128, 0, stream>>>(alpha, As, U, Mall, vall);
    kB_chunkscan<<<dim3(B_),         256, 0, stream>>>(Mall, vall, hstart, hT);
    kC_replay  <<<dim3(B_ * C_),     256, 0, stream>>>(alpha, As, U, hstart, hist);
    k3_y       <<<dim3(BT_ / 16, 4), 128, 0, stream>>>(x, alpha, Cs, Ds, hist, y);
}